// ChaoticRNN_53266184405811
// MI455X (gfx1250) — compile-verified
//
#include <hip/hip_runtime.h>
#include <stdint.h>

// ---------------- problem constants (match reference) ----------------
#define IN_F   1024
#define ASS    100000
#define OUT_F  1024
#define TOTAL  (IN_F + ASS + OUT_F)
#define NUM_ROUNDS 24

#define NT     256                // threads per block (8 wave32 waves)
#define TILE   2048               // edges staged per LDS tile
#define CHUNKS 2                  // 2 x B128 per thread per array = 2048 ints

// ---------------- CDNA5 async global->LDS staging ----------------
// GVS form: lds_dst_vgpr, 32-bit byte-offset vgpr, 64-bit sgpr base.
__device__ __forceinline__ void async_b128(uint32_t lds_off, uint32_t goff,
                                           const void* base) {
  asm volatile("global_load_async_to_lds_b128 %0, %1, %2"
               :: "v"(lds_off), "v"(goff), "s"(base)
               : "memory");
}
__device__ __forceinline__ void wait_async_le6() {
  asm volatile("s_wait_asynccnt 6" ::: "memory");
}
__device__ __forceinline__ void wait_async_0() {
  asm volatile("s_wait_asynccnt 0" ::: "memory");
}

// Stage one tile (TILE edges of src/dst/w) into LDS: 6 async B128 per thread.
__device__ __forceinline__ void stage_tile(const int* gs, const int* gd,
                                           const float* gw, int base, int tid,
                                           int* ls, int* ld, float* lw) {
#pragma unroll
  for (int c = 0; c < CHUNKS; ++c) {
    const int eo = tid * 4 + c * (NT * 4);          // element offset in tile
    const uint32_t gb = (uint32_t)((base + eo) * 4); // byte offset from array base
    async_b128((uint32_t)(uintptr_t)(ls + eo), gb, gs);
    async_b128((uint32_t)(uintptr_t)(ld + eo), gb, gd);
    async_b128((uint32_t)(uintptr_t)(lw + eo), gb, gw);
  }
}

// ---------------- kernels ----------------
__global__ void zero_kernel(float* __restrict__ mem, float* __restrict__ delta,
                            float* __restrict__ tfire, int* __restrict__ status,
                            unsigned char* __restrict__ fireflag,
                            unsigned char* __restrict__ got) {
  const int i = blockIdx.x * NT + threadIdx.x;
  if (i >= TOTAL) return;
  mem[i] = 0.0f; delta[i] = 0.0f; tfire[i] = 0.0f;
  status[i] = 0; fireflag[i] = 0; got[i] = 0;
}

__global__ void input_edge_kernel(const float* __restrict__ x,
                                  const float* __restrict__ iw,
                                  const int* __restrict__ isrc,
                                  const int* __restrict__ idst,
                                  float* __restrict__ mem,
                                  unsigned char* __restrict__ got, int E) {
  const int e = blockIdx.x * NT + threadIdx.x;
  if (e >= E) return;
  const int d = idst[e];
  unsafeAtomicAdd(mem + d, x[isrc[e]] * iw[e]);  // native global_atomic_add_f32
  got[d] = 1;                                    // benign write race (all write 1)
}

// mode 0: after input phase  -> status = got ? 1 : 0, then precompute fire/tanh
// mode 1: between rounds     -> apply post-update, then precompute fire/tanh
__global__ void node_kernel(float* __restrict__ mem, float* __restrict__ delta,
                            float* __restrict__ tfire, int* __restrict__ status,
                            unsigned char* __restrict__ fireflag,
                            unsigned char* __restrict__ got, int mode) {
  const int i = blockIdx.x * NT + threadIdx.x;
  if (i >= TOTAL) return;
  float m = mem[i];
  int st;
  if (mode == 0) {
    st = got[i] ? 1 : 0;
  } else {
    const int fired = fireflag[i];
    const float dl = delta[i];
    m = (fired ? 0.0f : m) + dl;   // fired neurons erase memory, then add delta
    mem[i] = m;
    const int st_old = status[i];
    st = fired ? 2 : ((got[i] && st_old == 0) ? 1 : st_old);
  }
  status[i] = st;
  const bool fire = (st == 1) && (i >= IN_F) && (i < IN_F + ASS);
  fireflag[i] = fire ? 1 : 0;
  tfire[i] = fire ? tanhf(m) : 0.0f;   // pay tanh once per node, not per edge
  delta[i] = 0.0f;
  got[i] = 0;
}

// One propagation round over all assoc edges.
// Edge streams double-buffered through LDS via CDNA5 async-to-LDS DMA.
__global__ __launch_bounds__(NT) void edge_kernel(
    const int* __restrict__ esrc, const int* __restrict__ edst,
    const float* __restrict__ ew,
    const unsigned char* __restrict__ fireflag,
    const float* __restrict__ tfire,
    float* __restrict__ delta, unsigned char* __restrict__ got, int E) {
  __shared__ __align__(16) int   s_src[2][TILE];
  __shared__ __align__(16) int   s_dst[2][TILE];
  __shared__ __align__(16) float s_w[2][TILE];

  const int tid = threadIdx.x;
  const int n_tiles = E / TILE;
  const int E_full = n_tiles * TILE;

  int t = blockIdx.x;
  int buf = 0;
  if (t < n_tiles)
    stage_tile(esrc, edst, ew, t * TILE, tid, s_src[0], s_dst[0], s_w[0]);

  while (t < n_tiles) {
    const int tn = t + (int)gridDim.x;
    if (tn < n_tiles) {
      // prefetch next tile into the other buffer, then wait for current tile
      stage_tile(esrc, edst, ew, tn * TILE, tid,
                 s_src[buf ^ 1], s_dst[buf ^ 1], s_w[buf ^ 1]);
      wait_async_le6();   // oldest 6 ops (current tile) complete; 6 in flight
    } else {
      wait_async_0();
    }
    __syncthreads();      // make all waves' staged slices visible

#pragma unroll
    for (int k = 0; k < TILE / NT; ++k) {
      const int idx = tid + k * NT;
      const int s = s_src[buf][idx];
      if (fireflag[s]) {                 // 102KB byte array: cheap random filter
        const float v = tfire[s] * s_w[buf][idx];
        const int d = s_dst[buf][idx];
        unsafeAtomicAdd(delta + d, v);   // no-return f32 atomic (STOREcnt)
        got[d] = 1;
      }
    }
    __syncthreads();      // all reads of buf done before it is re-staged
    buf ^= 1;
    t = tn;
  }

  // tail edges (E not a multiple of TILE): block 0 handles directly
  if (blockIdx.x == 0) {
    for (int e = E_full + tid; e < E; e += NT) {
      const int s = esrc[e];
      if (fireflag[s]) {
        const int d = edst[e];
        unsafeAtomicAdd(delta + d, tfire[s] * ew[e]);
        got[d] = 1;
      }
    }
  }
}

__global__ void out_kernel(const float* __restrict__ mem, float* __restrict__ out) {
  const int i = blockIdx.x * NT + threadIdx.x;
  if (i >= OUT_F) return;
  out[i] = tanhf(mem[IN_F + ASS + i]);
}

// ---------------- launch ----------------
extern "C" void kernel_launch(void* const* d_in, const int* in_sizes, int n_in,
                              void* d_out, int out_size, void* d_ws, size_t ws_size,
                              hipStream_t stream) {
  const float* x    = (const float*)d_in[0];
  const float* iw   = (const float*)d_in[1];
  const float* aw   = (const float*)d_in[2];
  const int*   isrc = (const int*)d_in[3];
  const int*   idst = (const int*)d_in[4];
  const int*   asrc = (const int*)d_in[5];
  const int*   adst = (const int*)d_in[6];
  const int E_in = in_sizes[1];
  const int E_as = in_sizes[2];
  float* out = (float*)d_out;

  // workspace layout (~1.85 MB), fully re-initialized every call
  const int TOTAL_PAD = 102400;  // TOTAL rounded up to 1024
  float* mem   = (float*)d_ws;
  float* delta = mem + TOTAL_PAD;
  float* tfire = delta + TOTAL_PAD;
  int*   status = (int*)(tfire + TOTAL_PAD);
  unsigned char* fireflag = (unsigned char*)(status + TOTAL_PAD);
  unsigned char* got = fireflag + TOTAL_PAD;

  const dim3 nb((TOTAL + NT - 1) / NT);
  zero_kernel<<<nb, NT, 0, stream>>>(mem, delta, tfire, status, fireflag, got);
  input_edge_kernel<<<(E_in + NT - 1) / NT, NT, 0, stream>>>(x, iw, isrc, idst,
                                                             mem, got, E_in);
  node_kernel<<<nb, NT, 0, stream>>>(mem, delta, tfire, status, fireflag, got, 0);

  int n_tiles = E_as / TILE;
  int eg = (n_tiles + 1) / 2;           // ~2 tiles/block -> double-buffer pipeline
  if (eg < 1) eg = 1;
  if (eg > 4096) eg = 4096;

  for (int r = 0; r < NUM_ROUNDS; ++r) {
    edge_kernel<<<eg, NT, 0, stream>>>(asrc, adst, aw, fireflag, tfire, delta,
                                       got, E_as);
    node_kernel<<<nb, NT, 0, stream>>>(mem, delta, tfire, status, fireflag, got, 1);
  }
  out_kernel<<<(OUT_F + NT - 1) / NT, NT, 0, stream>>>(mem, out);
}